// YoloLoss_43319040147853
// MI455X (gfx1250) — compile-verified
//
#include <hip/hip_runtime.h>

// ---------------- problem constants ----------------
#define G_BLOCKS 1024
#define THREADS  256
#define WAVES    8          // 256 threads / wave32
#define KPL      2          // float4 per lane per tile per array
#define TILE_F4  (32*KPL)   // 64 float4 per wave-tile per array

constexpr int CH      = 106;                         // 5 + 101 channels
constexpr int CELLS   = 16384 * 6 * 6;               // 589824
constexpr int NELEM   = CELLS * CH;                  // 62,521,344 (div by 4)
constexpr int NF4     = NELEM / 4;                   // 15,630,336
constexpr int NTILES  = NF4 / TILE_F4;               // 244,224 (exact)
constexpr int WSTRIDE = G_BLOCKS * WAVES;            // 8192 waves

// generic pointer to a __shared__ object: low 32 bits are the workgroup-
// relative LDS byte offset (flat-aperture encoding), which is what the
// async-to-LDS VDST operand wants.
__device__ __forceinline__ unsigned lds_off(const void* p) {
  return (unsigned)(unsigned long long)p;
}
// CDNA5 async copy: global (GV mode, 64-bit vaddr) -> LDS, tracked by ASYNCcnt
__device__ __forceinline__ void async_ld_b128(unsigned lds, const void* g) {
  asm volatile("global_load_async_to_lds_b128 %0, %1, off"
               :: "v"(lds), "v"(g) : "memory");
}
#define WAIT_ASYNC(N) asm volatile("s_wait_asynccnt %0" :: "i"(N) : "memory")

// ---------------- stage 1: streaming partial sums ----------------
__global__ void __launch_bounds__(THREADS)
yolo_partial(const float* __restrict__ pred, const float* __restrict__ tgt,
             float* __restrict__ partial)
{
  __shared__ float4 sP[WAVES][2][TILE_F4];   // 16 KB
  __shared__ float4 sT[WAVES][2][TILE_F4];   // 16 KB
  __shared__ float  red[WAVES][5];

  const int tid  = threadIdx.x;
  const int lane = tid & 31;
  const int wv   = tid >> 5;
  const int wgid = blockIdx.x * WAVES + wv;  // global wave id (uniform per wave)

  const float4* p4g = (const float4*)pred;
  const float4* t4g = (const float4*)tgt;

  float axy = 0.f, awh = 0.f, aco = 0.f, acn = 0.f, acl = 0.f;

  int t = wgid, buf = 0;
  if (t < NTILES) {                          // prologue prefetch into buf 0
    int base = t * TILE_F4 + lane;
#pragma unroll
    for (int k = 0; k < KPL; k++) {
      async_ld_b128(lds_off(&sP[wv][0][lane + 32*k]), p4g + base + 32*k);
      async_ld_b128(lds_off(&sT[wv][0][lane + 32*k]), t4g + base + 32*k);
    }
  }
  for (; t < NTILES; t += WSTRIDE) {
    int tn = t + WSTRIDE;
    if (tn < NTILES) {                       // prefetch next tile into buf^1
      int nb = buf ^ 1, base = tn * TILE_F4 + lane;
#pragma unroll
      for (int k = 0; k < KPL; k++) {
        async_ld_b128(lds_off(&sP[wv][nb][lane + 32*k]), p4g + base + 32*k);
        async_ld_b128(lds_off(&sT[wv][nb][lane + 32*k]), t4g + base + 32*k);
      }
      WAIT_ASYNC(2 * KPL);                   // current tile's 4 copies done
    } else {
      WAIT_ASYNC(0);
    }
#pragma unroll
    for (int k = 0; k < KPL; k++) {
      float4 tv = sT[wv][buf][lane + 32*k];
      float4 pv = sP[wv][buf][lane + 32*k];
      unsigned g0   = (unsigned)(t * TILE_F4 + lane + 32*k) * 4u;
      unsigned cell = g0 / 106u;
      unsigned ch0  = g0 - cell * 106u;
      // object mask for this cell (and next cell if the float4 spans one);
      // streams through L2, essentially free
      float confA = tgt[cell * 106u + 4u];
      float confB = (ch0 > 102u) ? tgt[(cell + 1u) * 106u + 4u] : confA;
      float tc[4] = {tv.x, tv.y, tv.z, tv.w};
      float pc[4] = {pv.x, pv.y, pv.z, pv.w};
#pragma unroll
      for (int c = 0; c < 4; c++) {
        unsigned ch = ch0 + (unsigned)c;
        float conf  = confA;
        if (ch >= 106u) { ch -= 106u; conf = confB; }
        float obj = (conf != 0.0f) ? 1.0f : 0.0f;
        float tvc = tc[c], pvc = pc[c];
        float d = tvc - pvc, d2 = d * d;
        if (ch < 2u) {
          axy += obj * d2;
        } else if (ch < 4u) {
          float s = sqrtf(tvc) - sqrtf(fmaxf(pvc, 0.0f));
          awh += obj * (s * s);
        } else if (ch == 4u) {
          aco += obj * d2;
          acn += d2 - obj * d2;              // (1-obj)*d2
        } else {
          acl += obj * d2;
        }
      }
    }
    buf ^= 1;
  }

  // deterministic wave32 reduce
#pragma unroll
  for (int off = 16; off > 0; off >>= 1) {
    axy += __shfl_xor(axy, off, 32);
    awh += __shfl_xor(awh, off, 32);
    aco += __shfl_xor(aco, off, 32);
    acn += __shfl_xor(acn, off, 32);
    acl += __shfl_xor(acl, off, 32);
  }
  if (lane == 0) {
    red[wv][0] = axy; red[wv][1] = awh; red[wv][2] = aco;
    red[wv][3] = acn; red[wv][4] = acl;
  }
  __syncthreads();
  if (tid < 5) {
    float s = 0.f;
#pragma unroll
    for (int w = 0; w < WAVES; w++) s += red[w][tid];
    partial[tid * G_BLOCKS + blockIdx.x] = s;   // [5][G_BLOCKS]
  }
}

// ---------------- stage 2: final reduce + WMMA combine ----------------
typedef float v2f __attribute__((ext_vector_type(2)));
typedef float v8f __attribute__((ext_vector_type(8)));

__global__ void __launch_bounds__(32)
yolo_final(const float* __restrict__ partial, float* __restrict__ out)
{
  const int lane = threadIdx.x;
  float s[5];
#pragma unroll
  for (int j = 0; j < 5; j++) {
    float v = 0.f;
    for (int i = lane; i < G_BLOCKS; i += 32) v += partial[j * G_BLOCKS + i];
#pragma unroll
    for (int off = 16; off > 0; off >>= 1) v += __shfl_xor(v, off, 32);
    s[j] = v;                                // full sum in every lane
  }
  const float invB = 1.0f / 16384.0f;        // exact power-of-two scale

  // D = A(16x4) * B(4x16) + C : place weights {5,5,1,0.5} in row 0 of A,
  // sums {sxy,sco,swh,scn} in column 0 of B, scls in C[0][0].
  // A layout: lanes0-15 hold K=0(v0),K=1(v1); lanes16-31 hold K=2(v0),K=3(v1)
  // B layout: v0 = rows0/2 (lane halves), v1 = rows1/3
  v2f A, Bm; v8f Cm = {};
  A[0]  = (lane == 0) ? 5.0f : ((lane == 16) ? 1.0f : 0.0f);  // A[0][0], A[0][2]
  A[1]  = (lane == 0) ? 5.0f : ((lane == 16) ? 0.5f : 0.0f);  // A[0][1], A[0][3]
  Bm[0] = (lane == 0) ? s[0] : ((lane == 16) ? s[1] : 0.0f);  // B[0][0], B[2][0]
  Bm[1] = (lane == 0) ? s[2] : ((lane == 16) ? s[3] : 0.0f);  // B[1][0], B[3][0]
  Cm[0] = (lane == 0) ? s[4] : 0.0f;                          // C[0][0]
  // EXEC is all ones here (single fully-active wave, uniform control flow)
  v8f D = __builtin_amdgcn_wmma_f32_16x16x4_f32(
      false, A, false, Bm, (short)0, Cm, false, false);

  if (lane == 0) {
    out[0] = s[0] * invB;                    // loss_xy
    out[1] = s[1] * invB;                    // loss_wh
    out[2] = s[2] * invB;                    // loss_conf_obj
    out[3] = s[3] * invB;                    // loss_conf_noobj
    out[4] = s[4] * invB;                    // loss_class
    out[5] = D[0] * (invB * invB);           // combined loss (/B twice)
  }
}

// ---------------- launch ----------------
extern "C" void kernel_launch(void* const* d_in, const int* in_sizes, int n_in,
                              void* d_out, int out_size, void* d_ws, size_t ws_size,
                              hipStream_t stream) {
  const float* pred = (const float*)d_in[0];
  const float* tgt  = (const float*)d_in[1];
  float* partial    = (float*)d_ws;          // 5 * G_BLOCKS floats (20 KB)
  yolo_partial<<<G_BLOCKS, THREADS, 0, stream>>>(pred, tgt, partial);
  yolo_final<<<1, 32, 0, stream>>>(partial, (float*)d_out);
}